// GraphSAGEClassifier_6923487282665
// MI455X (gfx1250) — compile-verified
//
#include <hip/hip_runtime.h>

#define N_NODES   50000
#define N_EDGES   800000
#define NUM_GRAPHS 256
#define D         128

typedef __attribute__((ext_vector_type(2))) float v2f;
typedef __attribute__((ext_vector_type(8))) float v8f;

// ---------------------------------------------------------------- utilities
__global__ __launch_bounds__(256) void zero_f4(float* __restrict__ p, int n4) {
    int i = blockIdx.x * 256 + threadIdx.x;
    if (i < n4) reinterpret_cast<float4*>(p)[i] = make_float4(0.f, 0.f, 0.f, 0.f);
}

__global__ __launch_bounds__(256) void deg_count(const int* __restrict__ dst,
                                                 float* __restrict__ deg) {
    int e = blockIdx.x * 256 + threadIdx.x;
    if (e < N_EDGES) atomicAdd(&deg[dst[e]], 1.0f);
}

__global__ __launch_bounds__(256) void deg_invert(float* __restrict__ deg) {
    int i = blockIdx.x * 256 + threadIdx.x;
    if (i < N_NODES) deg[i] = 1.0f / fmaxf(deg[i], 1.0f);
}

// ------------------------------------------------------------ edge scatter
// one lane = (edge, 4-feature chunk); float4 gather, 4 atomic f32 adds
__global__ __launch_bounds__(256) void scatter_accum(const float* __restrict__ h,
                                                     const int* __restrict__ src,
                                                     const int* __restrict__ dst,
                                                     float* __restrict__ agg) {
    long long tid = (long long)blockIdx.x * 256 + threadIdx.x;
    int e = (int)(tid >> 5);
    int c = ((int)tid & 31) * 4;
    if (e >= N_EDGES) return;
    int s = src[e], d = dst[e];
    float4 v = reinterpret_cast<const float4*>(h + (size_t)s * D)[c >> 2];
    float* a = agg + (size_t)d * D + c;
    atomicAdd(a + 0, v.x);
    atomicAdd(a + 1, v.y);
    atomicAdd(a + 2, v.z);
    atomicAdd(a + 3, v.w);
}

// ------------------------------------------------- fused SAGE layer (WMMA)
// out[m][n] = relu( (agg[m]*inv_deg[m]) . Wl[n] + bl[n] + h[m] . Wr[n] )
// block = 256 thr = 8 wave32; block tile = 16 nodes x 128 feats;
// wave w owns output features [16w, 16w+16); two 32-step f32 WMMA chains.
__global__ __launch_bounds__(256) void sage_wmma(const float* __restrict__ agg,
                                                 const float* __restrict__ h,
                                                 const float* __restrict__ inv_deg,
                                                 const float* __restrict__ Wl,
                                                 const float* __restrict__ bl,
                                                 const float* __restrict__ Wr,
                                                 float* __restrict__ out) {
    __shared__ __align__(16) float sA[16][D];  // agg * inv_deg
    __shared__ __align__(16) float sH[16][D];  // root features

    const int m0 = blockIdx.x * 16;
    const int t  = threadIdx.x;

    // stage 16x128 tiles (512 float4 each), coalesced
    for (int i = t; i < 512; i += 256) {
        int m = i >> 5, c4 = i & 31;
        float s = inv_deg[m0 + m];
        float4 va = reinterpret_cast<const float4*>(agg + (size_t)(m0 + m) * D)[c4];
        va.x *= s; va.y *= s; va.z *= s; va.w *= s;
        *reinterpret_cast<float4*>(&sA[m][c4 * 4]) = va;
        *reinterpret_cast<float4*>(&sH[m][c4 * 4]) =
            reinterpret_cast<const float4*>(h + (size_t)(m0 + m) * D)[c4];
    }
    __syncthreads();

    const int lane = t & 31;
    const int wv   = t >> 5;      // 0..7 -> N tile
    const int half = lane >> 4;   // 0: K pair (k,k+1); 1: K pair (k+2,k+3)
    const int mn   = lane & 15;   // M row for A, N col for B/C/D
    const int n    = wv * 16 + mn;

    v8f c = {};
#pragma unroll
    for (int k0 = 0; k0 < D; k0 += 4) {
        int k = k0 + 2 * half;
        v2f a = { sA[mn][k], sA[mn][k + 1] };
        v2f b = { Wl[(size_t)n * D + k], Wl[(size_t)n * D + k + 1] };
        c = __builtin_amdgcn_wmma_f32_16x16x4_f32(false, a, false, b,
                                                  (short)0, c, false, false);
    }
#pragma unroll
    for (int k0 = 0; k0 < D; k0 += 4) {
        int k = k0 + 2 * half;
        v2f a = { sH[mn][k], sH[mn][k + 1] };
        v2f b = { Wr[(size_t)n * D + k], Wr[(size_t)n * D + k + 1] };
        c = __builtin_amdgcn_wmma_f32_16x16x4_f32(false, a, false, b,
                                                  (short)0, c, false, false);
    }

    const float bias = bl[n];
#pragma unroll
    for (int v = 0; v < 8; ++v) {
        int row = m0 + v + 8 * half;           // C/D layout: M = vgpr + 8*half
        out[(size_t)row * D + n] = fmaxf(c[v] + bias, 0.0f);
    }
}

// ---------------------------------------------------------------- pooling
__global__ __launch_bounds__(256) void pool_accum(const float* __restrict__ h,
                                                  const int* __restrict__ batch,
                                                  float* __restrict__ sums,
                                                  float* __restrict__ cnt) {
    long long tid = (long long)blockIdx.x * 256 + threadIdx.x;
    int nidx = (int)(tid >> 5);
    int c = ((int)tid & 31) * 4;
    if (nidx >= N_NODES) return;
    int g = batch[nidx];
    float4 v = reinterpret_cast<const float4*>(h + (size_t)nidx * D)[c >> 2];
    float* s = sums + (size_t)g * D + c;
    atomicAdd(s + 0, v.x);
    atomicAdd(s + 1, v.y);
    atomicAdd(s + 2, v.z);
    atomicAdd(s + 3, v.w);
    if (c == 0) atomicAdd(&cnt[g], 1.0f);
}

// --------------------------------------------------------------- MLP head
__global__ __launch_bounds__(64) void mlp_head(const float* __restrict__ sums,
                                               const float* __restrict__ cnt,
                                               const float* __restrict__ Wc1,
                                               const float* __restrict__ bc1,
                                               const float* __restrict__ Wc2,
                                               const float* __restrict__ bc2,
                                               float* __restrict__ out) {
    __shared__ float g[D];
    __shared__ float hid[64];
    const int gi = blockIdx.x;
    const int j  = threadIdx.x;

    float inv = 1.0f / fmaxf(cnt[gi], 1.0f);
    for (int i = j; i < D; i += 64) g[i] = sums[(size_t)gi * D + i] * inv;
    __syncthreads();

    float acc = bc1[j];
#pragma unroll 4
    for (int k = 0; k < D; ++k) acc += g[k] * Wc1[(size_t)j * D + k];
    hid[j] = fmaxf(acc, 0.0f);
    __syncthreads();

    if (j < 2) {
        float acc2 = bc2[j];
#pragma unroll 4
        for (int k = 0; k < 64; ++k) acc2 += hid[k] * Wc2[j * 64 + k];
        out[gi * 2 + j] = acc2;
    }
}

// ------------------------------------------------------------------ launch
extern "C" void kernel_launch(void* const* d_in, const int* in_sizes, int n_in,
                              void* d_out, int out_size, void* d_ws, size_t ws_size,
                              hipStream_t stream) {
    const float* x   = (const float*)d_in[0];
    const int*   ei  = (const int*)d_in[1];
    const int*   bat = (const int*)d_in[2];
    const float* W1l = (const float*)d_in[3];
    const float* b1l = (const float*)d_in[4];
    const float* W1r = (const float*)d_in[5];
    const float* W2l = (const float*)d_in[6];
    const float* b2l = (const float*)d_in[7];
    const float* W2r = (const float*)d_in[8];
    const float* W3l = (const float*)d_in[9];
    const float* b3l = (const float*)d_in[10];
    const float* W3r = (const float*)d_in[11];
    const float* Wc1 = (const float*)d_in[12];
    const float* bc1 = (const float*)d_in[13];
    const float* Wc2 = (const float*)d_in[14];
    const float* bc2 = (const float*)d_in[15];

    const int* src = ei;
    const int* dst = ei + N_EDGES;

    // workspace layout (floats): ~77.1 MB total
    float* ws      = (float*)d_ws;
    float* inv_deg = ws;                                 // 50000 (pad to 50048)
    float* agg     = ws + 50048;                         // 6.4M
    float* h1      = agg + (size_t)N_NODES * D;          // 6.4M
    float* h2      = h1  + (size_t)N_NODES * D;          // 6.4M
    float* sums    = h2  + (size_t)N_NODES * D;          // 32768
    float* cnt     = sums + NUM_GRAPHS * D;              // 256 (contig w/ sums)

    const int featN4        = N_NODES * D / 4;           // 1.6M float4
    const int scatterBlocks = N_EDGES * 32 / 256;        // 100000
    const int gemmBlocks    = N_NODES / 16;              // 3125 (exact)

    // degrees -> inv_deg
    zero_f4<<<(N_NODES / 4 + 255) / 256, 256, 0, stream>>>(inv_deg, N_NODES / 4);
    deg_count<<<(N_EDGES + 255) / 256, 256, 0, stream>>>(dst, inv_deg);
    deg_invert<<<(N_NODES + 255) / 256, 256, 0, stream>>>(inv_deg);

    // layer 1: x -> h1
    zero_f4<<<(featN4 + 255) / 256, 256, 0, stream>>>(agg, featN4);
    scatter_accum<<<scatterBlocks, 256, 0, stream>>>(x, src, dst, agg);
    sage_wmma<<<gemmBlocks, 256, 0, stream>>>(agg, x, inv_deg, W1l, b1l, W1r, h1);

    // layer 2: h1 -> h2
    zero_f4<<<(featN4 + 255) / 256, 256, 0, stream>>>(agg, featN4);
    scatter_accum<<<scatterBlocks, 256, 0, stream>>>(h1, src, dst, agg);
    sage_wmma<<<gemmBlocks, 256, 0, stream>>>(agg, h1, inv_deg, W2l, b2l, W2r, h2);

    // layer 3: h2 -> h1
    zero_f4<<<(featN4 + 255) / 256, 256, 0, stream>>>(agg, featN4);
    scatter_accum<<<scatterBlocks, 256, 0, stream>>>(h2, src, dst, agg);
    sage_wmma<<<gemmBlocks, 256, 0, stream>>>(agg, h2, inv_deg, W3l, b3l, W3r, h1);

    // global mean pool + head
    int poolZero4 = (NUM_GRAPHS * D + NUM_GRAPHS) / 4;   // sums + cnt contiguous
    zero_f4<<<(poolZero4 + 255) / 256, 256, 0, stream>>>(sums, poolZero4);
    pool_accum<<<N_NODES * 32 / 256, 256, 0, stream>>>(h1, bat, sums, cnt);
    mlp_head<<<NUM_GRAPHS, 64, 0, stream>>>(sums, cnt, Wc1, bc1, Wc2, bc2,
                                            (float*)d_out);
}